// SimpleEnergyModel_43903155700342
// MI455X (gfx1250) — compile-verified
//
#include <hip/hip_runtime.h>
#include <hip/hip_bf16.h>
#include <float.h>
#include <math.h>

// ---------------------------------------------------------------------------
// Reference collapses to:
//   S     = sum_{i,j} |c_i - c_j|^2  = 2*N*sum|c_i|^2 - 2*|sum c_i|^2   (scalar!)
//   r     = nan_to_num(1/sqrt(S))
//   Wsum  = sum_{a,b} cnt[a]*cnt[b]*weights[a*(a+1)/2 + b]   (100x100, padded 128)
//   out   = Wsum * r + bias
// Kernel A: O(N) reduction (sums + ds_add_f32 histogram) -> d_ws
// Kernel B: quadratic form cnt^T * W * cnt via V_WMMA_F32_16X16X4_F32,
//           branch-free masked loads (clamp + cndmask), fully unrolled K loop.
// ---------------------------------------------------------------------------

typedef __attribute__((ext_vector_type(2))) float v2f;
typedef __attribute__((ext_vector_type(8))) float v8f;

#define N_ATOMS    4096
#define NBINS      100
#define N_WEIGHTS  5050          // NBINS*(NBINS+1)/2
#define PADN       128
// d_ws float layout: [0] = reciprocal_d, [8..8+127] = padded counts (as float)
#define WS_RECIP   0
#define WS_CNT     8

__global__ void __launch_bounds__(1024)
energy_reduce_kernel(const float* __restrict__ coords,
                     const int*   __restrict__ atom_ix,
                     float* __restrict__ ws, int n)
{
    __shared__ float s_sum[4];          // sx, sy, sz, s|c|^2
    __shared__ float s_cnt[PADN];       // histogram, float

    const int t = threadIdx.x;
    if (t < PADN) s_cnt[t] = 0.0f;
    if (t < 4)    s_sum[t] = 0.0f;
    __syncthreads();

    float sx = 0.f, sy = 0.f, sz = 0.f, sq = 0.f;
    for (int i = t; i < n; i += blockDim.x) {
        const float x = coords[3 * i + 0];
        const float y = coords[3 * i + 1];
        const float z = coords[3 * i + 2];
        sx += x; sy += y; sz += z;
        sq = fmaf(x, x, sq); sq = fmaf(y, y, sq); sq = fmaf(z, z, sq);
        const int a = atom_ix[i];                // 0..99
        atomicAdd(&s_cnt[a], 1.0f);             // ds_add_f32
    }
    atomicAdd(&s_sum[0], sx);
    atomicAdd(&s_sum[1], sy);
    atomicAdd(&s_sum[2], sz);
    atomicAdd(&s_sum[3], sq);
    __syncthreads();

    if (t < PADN) ws[WS_CNT + t] = s_cnt[t];    // bins >= NBINS stay 0
    if (t == 0) {
        const float fn = (float)n;
        const float S  = 2.0f * fn * s_sum[3]
                       - 2.0f * (s_sum[0]*s_sum[0] + s_sum[1]*s_sum[1] + s_sum[2]*s_sum[2]);
        const float d  = sqrtf(S);
        float r = 1.0f / d;
        // match jnp.nan_to_num(x, nan=0.0): nan->0, +inf->FLT_MAX, -inf->-FLT_MAX
        if (isnan(r))        r = 0.0f;
        else if (isinf(r))   r = (r > 0.0f) ? FLT_MAX : -FLT_MAX;
        ws[WS_RECIP] = r;
    }
}

// 8 waves (256 threads), wave i owns the 16-row I-tile i of the padded 128x128 W.
// For each 4-wide K chunk: D += A(16x4) * B(4x16), B broadcasting v[k] across all
// 16 columns -> every column of D accumulates y_I = (W*v)_I.
// 32-bit A 16x4 layout (ISA 7.12.2): lanes 0-15 hold K={0,1}, lanes 16-31 hold
// K={2,3} in VGPR0/VGPR1; B 4x16 mirrored. Broadcast makes B layout-robust.
// Loads are branch-free: index clamped in-range, zero-padding via cndmask.
__global__ void __launch_bounds__(256)
energy_wmma_kernel(const float* __restrict__ weights,
                   const float* __restrict__ bias,
                   const float* __restrict__ ws,
                   float* __restrict__ out)
{
    __shared__ float s_v[PADN];
    __shared__ float s_tot;

    const int t = threadIdx.x;
    if (t < PADN) s_v[t] = ws[WS_CNT + t];
    if (t == 0)   s_tot = 0.0f;
    __syncthreads();

    const int  wave = t >> 5;        // 0..7  (wave32)
    const int  lane = t & 31;
    const int  m    = lane & 15;     // row within tile
    const int  hi   = lane >> 4;     // 0 or 1 -> K half of the 4-chunk
    const int  a    = wave * 16 + m; // global row 0..127
    const int  tri  = (a * (a + 1)) >> 1;
    const bool arow = (a < NBINS);

    // Pull this wave's weight row toward the caches (global_prefetch_b8).
    __builtin_prefetch(weights + (arow ? tri : 0), 0, 0);

    v8f acc = {};
    #pragma unroll
    for (int kb = 0; kb < PADN; kb += 4) {
        const int k0 = kb + 2 * hi;
        const int k1 = k0 + 1;
        // Unconditional, in-range loads (clamp), padding applied via select:
        int i0 = tri + k0; i0 = (i0 > N_WEIGHTS - 1) ? (N_WEIGHTS - 1) : i0;
        int i1 = tri + k1; i1 = (i1 > N_WEIGHTS - 1) ? (N_WEIGHTS - 1) : i1;
        const float w0 = weights[i0];
        const float w1 = weights[i1];
        v2f A, B;
        A.x = (arow && (k0 < NBINS)) ? w0 : 0.0f;   // k0<NBINS: compile-time
        A.y = (arow && (k1 < NBINS)) ? w1 : 0.0f;   // after full unroll
        B.x = s_v[k0];               // broadcast across all 16 columns
        B.y = s_v[k1];
        acc = __builtin_amdgcn_wmma_f32_16x16x4_f32(
            /*neg_a=*/false, A, /*neg_b=*/false, B,
            /*c_mod=*/(short)0, acc, /*reuse_a=*/false, /*reuse_b=*/false);
    }

    // C/D layout: VGPR r, lane L -> M = r + 8*(L>=16), N = L%16; every column == y_I.
    // Sum over all lanes/regs of v[M]*acc[r] == 16 * (v_I . y_I).
    float local = 0.0f;
    #pragma unroll
    for (int r = 0; r < 8; ++r)
        local += s_v[wave * 16 + hi * 8 + r] * acc[r];

    atomicAdd(&s_tot, local);        // ds_add_f32
    __syncthreads();

    if (t == 0)
        out[0] = (s_tot * (1.0f / 16.0f)) * ws[WS_RECIP] + bias[0];
}

extern "C" void kernel_launch(void* const* d_in, const int* in_sizes, int n_in,
                              void* d_out, int out_size, void* d_ws, size_t ws_size,
                              hipStream_t stream)
{
    const float* coords  = (const float*)d_in[0];   // (4096, 3) f32
    const int*   atom_ix = (const int*)  d_in[1];   // (4096,)   int
    const float* weights = (const float*)d_in[2];   // (5050,)   f32
    const float* bias    = (const float*)d_in[3];   // (1,)      f32
    float*       out     = (float*)d_out;
    float*       ws      = (float*)d_ws;            // needs (8+128)*4 = 544 bytes

    const int n = in_sizes[1];                      // N_ATOMS

    energy_reduce_kernel<<<1, 1024, 0, stream>>>(coords, atom_ix, ws, n);
    energy_wmma_kernel  <<<1, 256,  0, stream>>>(weights, bias, ws, out);
}